// GCN_29764123361867
// MI455X (gfx1250) — compile-verified
//
#include <hip/hip_runtime.h>

typedef __attribute__((ext_vector_type(2))) float v2f;
typedef __attribute__((ext_vector_type(8))) float v8f;

#define FEATS 128   // IN_FEATS == OUT_FEATS == 128

// ---------------- kernel 1: zero the aggregation buffer ----------------
__global__ void gcn_zero(float4* __restrict__ p, int n4) {
  int i = blockIdx.x * blockDim.x + threadIdx.x;
  if (i < n4) p[i] = make_float4(0.f, 0.f, 0.f, 0.f);
}

// ------------- kernel 2: edge gather + scatter-add (segment_sum) -------------
// One wave32 per edge: lane l moves feature[src][4l..4l+3] -> agg[dst][4l..4l+3]
// via float4 load + 4 native global_atomic_add_f32 (agg fits in L2).
__global__ __launch_bounds__(256) void gcn_scatter(const float* __restrict__ feat,
                                                   const int* __restrict__ src,
                                                   const int* __restrict__ dst,
                                                   float* __restrict__ agg,
                                                   int nEdges) {
  int t = blockIdx.x * blockDim.x + threadIdx.x;
  int edge = t >> 5;
  int lane = t & 31;
  if (edge >= nEdges) return;
  int s = src[edge];
  int d = dst[edge];
  float4 v = ((const float4*)(feat + (size_t)s * FEATS))[lane];
  float* o = agg + (size_t)d * FEATS + lane * 4;
  unsafeAtomicAdd(o + 0, v.x);
  unsafeAtomicAdd(o + 1, v.y);
  unsafeAtomicAdd(o + 2, v.z);
  unsafeAtomicAdd(o + 3, v.w);
}

// ------------- kernel 3: fp32 WMMA GEMM + bias + ReLU -------------
// h = relu(agg @ W^T + b); one wave per 16x16 output tile,
// K=128 swept in 32 steps of v_wmma_f32_16x16x4_f32.
__global__ __launch_bounds__(256) void gcn_gemm(const float* __restrict__ agg,
                                                const float* __restrict__ W,
                                                const float* __restrict__ bias,
                                                float* __restrict__ out,
                                                int mTiles) {
  const int nTiles = FEATS / 16;              // 8 tiles across output features
  int wave = (blockIdx.x * blockDim.x + threadIdx.x) >> 5;   // global wave id
  int lane = threadIdx.x & 31;
  int mt = wave / nTiles;                     // wave-uniform
  int nt = wave % nTiles;
  if (mt >= mTiles) return;                   // wave-uniform exit: EXEC all-1s inside

  const int m0 = mt * 16;
  const int n0 = nt * 16;
  const int row   = lane & 15;                // A row / B col / C col
  const int khalf = lane >> 4;                // 0: K{0,1}, 1: K{2,3} within each step

  // A fragment source: agg[m0+row][k + khalf*2 + {0,1}]
  const float* aBase = agg + (size_t)(m0 + row) * FEATS + khalf * 2;
  // B fragment source: B[k][n] = W[n][k] -> W[(n0+col)][k + khalf*2 + {0,1}]
  const float* bBase = W + (size_t)(n0 + row) * FEATS + khalf * 2;

  v8f c = {};
#pragma unroll
  for (int k = 0; k < FEATS; k += 4) {
    v2f a, b;
    a.x = aBase[k];
    a.y = aBase[k + 1];
    b.x = bBase[k];
    b.y = bBase[k + 1];
    // 8 args: (neg_a, A, neg_b, B, c_mod, C, reuse_a, reuse_b)
    c = __builtin_amdgcn_wmma_f32_16x16x4_f32(false, a, false, b,
                                              (short)0, c, false, false);
  }

  float bv = bias[n0 + row];                  // lane's output column
#pragma unroll
  for (int r = 0; r < 8; ++r) {
    int orow = m0 + r + khalf * 8;            // C/D layout: VGPR r -> M = r + 8*khalf
    float v = c[r] + bv;
    out[(size_t)orow * FEATS + n0 + row] = v > 0.f ? v : 0.f;
  }
}

extern "C" void kernel_launch(void* const* d_in, const int* in_sizes, int n_in,
                              void* d_out, int out_size, void* d_ws, size_t ws_size,
                              hipStream_t stream) {
  const float* feature = (const float*)d_in[0];   // [N, 128]
  const int*   src_idx = (const int*)d_in[1];     // [E]
  const int*   dst_idx = (const int*)d_in[2];     // [E]
  const float* W       = (const float*)d_in[3];   // [128, 128] row-major [out,in]
  const float* bvec    = (const float*)d_in[4];   // [128]
  float*       out     = (float*)d_out;           // [N, 128]

  const int nNodes = in_sizes[0] / FEATS;
  const int nEdges = in_sizes[1];

  float* agg = (float*)d_ws;                      // N*128 floats scratch (25.6 MB)

  // 1) agg = 0
  {
    int n4 = nNodes * FEATS / 4;
    int blocks = (n4 + 255) / 256;
    gcn_zero<<<blocks, 256, 0, stream>>>((float4*)agg, n4);
  }

  // 2) agg[dst] += feature[src] over edges (one wave per edge)
  {
    long long threads = (long long)nEdges * 32;
    int blocks = (int)((threads + 255) / 256);
    gcn_scatter<<<blocks, 256, 0, stream>>>(feature, src_idx, dst_idx, agg, nEdges);
  }

  // 3) out = relu(agg @ W^T + b) via fp32 WMMA
  {
    int mTiles = (nNodes + 15) / 16;                // 3125 for N=50000
    int tiles = mTiles * (FEATS / 16);              // one wave per tile
    int blocks = (tiles + 7) / 8;                   // 8 waves per 256-thread block
    gcn_gemm<<<blocks, 256, 0, stream>>>(agg, W, bvec, out, mTiles);
  }
}